// DQC_50259707297806
// MI455X (gfx1250) — compile-verified
//
#include <hip/hip_runtime.h>
#include <cmath>

// ---------------------------------------------------------------------------
// 16-qubit state-vector simulator for the DQC reference, reformulated as
// 16x16 complex GEMMs over 4-qubit blocks so the inner loop is
// V_WMMA_F32_16X16X4_F32 on gfx1250 (wave32).
//
// Bit convention (from the reference's reshape): wire q <-> bit (15-q) of the
// amplitude index n.  Blocks: blk0 = bits[15:12] (wires 0-3), blk1 = bits[11:8],
// blk2 = bits[7:4], blk3 = bits[3:0].
//
// Per layer:  G_blk = P(w1,w2) * P(w0,w1) * P(w2,w3) * Kron(U_w0..U_w3)
// (within-block CNOTs folded in).  Cross-block CNOTs (3,4),(7,8),(11,12) are
// bit-conditioned permutations folded into tile load/store addressing and the
// readout weight.
// ---------------------------------------------------------------------------

typedef float v2f __attribute__((ext_vector_type(2)));
typedef float v8f __attribute__((ext_vector_type(8)));

static __device__ __forceinline__ v8f wmma_f32(v2f a, v2f b, v8f c) {
  // D(16x16) = A(16x4) * B(4x16) + C   (f32, wave32)
  return __builtin_amdgcn_wmma_f32_16x16x4_f32(
      /*neg_a=*/false, a, /*neg_b=*/false, b,
      /*c_mod=*/(short)0, c, /*reuse_a=*/false, /*reuse_b=*/false);
}

struct Cf { float r, i; };

// WMMA B-operand fragments for one gate (Re, Im, -Im), all 4 K-steps.
struct GateFrag { v2f br[4], bi[4], bn[4]; };

static __device__ __forceinline__ GateFrag load_gate(const float* __restrict__ gw,
                                                     int lane) {
  GateFrag f;
  const int j     = lane & 15;
  const int khalf = (lane >> 4) << 1;   // lanes 0-15: K=k0,k0+1; 16-31: K=k0+2,k0+3
  const float* Br = gw;
  const float* Bi = gw + 256;
  const float* Bn = gw + 512;
#pragma unroll
  for (int ks = 0; ks < 4; ++ks) {
    const int s0 = ks * 4 + khalf;
    f.br[ks].x = Br[s0 * 16 + j]; f.br[ks].y = Br[(s0 + 1) * 16 + j];
    f.bi[ks].x = Bi[s0 * 16 + j]; f.bi[ks].y = Bi[(s0 + 1) * 16 + j];
    f.bn[ks].x = Bn[s0 * 16 + j]; f.bn[ks].y = Bn[(s0 + 1) * 16 + j];
  }
  return f;
}

// ---------------------------------------------------------------------------
// Kernel 1: build the 32 fused layer-block gates (8 layers x 4 blocks).
// Stored directly in WMMA-B orientation: B[s][j] = G[j][s], three 16x16 f32
// planes per gate: Re, Im, -Im  (768 floats per (l,blk)).
// ---------------------------------------------------------------------------
__global__ void __launch_bounds__(256)
dqc_build_gates(const float* __restrict__ params, float* __restrict__ gates) {
  const int l   = blockIdx.x;   // layer 0..7
  const int blk = blockIdx.y;   // block 0..3
  const int t   = threadIdx.x;  // one (r,s) entry per thread
  const int r   = t >> 4;
  const int s   = t & 15;

  // Per-wire U = RZ(p2) * RY(p1) * RZ(p0)
  Cf u[4][2][2];
#pragma unroll
  for (int j = 0; j < 4; ++j) {
    const int q = blk * 4 + j;                 // wire; maps to block bit (3-j)
    const float a  = params[(l * 3 + 0) * 16 + q];
    const float bb = params[(l * 3 + 1) * 16 + q];
    const float c  = params[(l * 3 + 2) * 16 + q];
    const float cb = cosf(0.5f * bb), sb = sinf(0.5f * bb);
    const float apc = 0.5f * (a + c), amc = 0.5f * (a - c);
    const float ca = cosf(apc), sa = sinf(apc);
    const float cm = cosf(amc), sm = sinf(amc);
    u[j][0][0] = { cb * ca, -cb * sa};   //  cb * e^{-i(a+c)/2}
    u[j][0][1] = {-sb * cm, -sb * sm};   // -sb * e^{+i(a-c)/2}
    u[j][1][0] = { sb * cm, -sb * sm};   //  sb * e^{-i(a-c)/2}
    u[j][1][1] = { cb * ca,  cb * sa};   //  cb * e^{+i(a+c)/2}
  }

  // Row permutation from within-block CNOTs: (w0,w1),(w2,w3) then (w1,w2).
  // Block-local bits: w0->bit3, w1->bit2, w2->bit1, w3->bit0.
  int rp = r ^ ((((r  >> 2) & 1)) << 1);   // CNOT(w1,w2): ctrl bit2 -> flip bit1
  rp    ^= ((((rp >> 3) & 1)) << 2);       // CNOT(w0,w1): ctrl bit3 -> flip bit2
  rp    ^= (((rp >> 1) & 1));              // CNOT(w2,w3): ctrl bit1 -> flip bit0

  // K[rp, s] = prod_j U_wj[bit_{3-j}(rp), bit_{3-j}(s)]
  float kr = 1.0f, ki = 0.0f;
#pragma unroll
  for (int j = 0; j < 4; ++j) {
    const int rb = (rp >> (3 - j)) & 1;
    const int sc = (s  >> (3 - j)) & 1;
    const Cf e = u[j][rb][sc];
    const float nr = kr * e.r - ki * e.i;
    ki = kr * e.i + ki * e.r;
    kr = nr;
  }

  float* g = gates + (l * 4 + blk) * 768;
  g[      s * 16 + r] =  kr;   // B-real   (B[s][j=r] = G[r][s])
  g[256 + s * 16 + r] =  ki;   // B-imag
  g[512 + s * 16 + r] = -ki;   // B-imag negated (for the real accumulator)
}

// ---------------------------------------------------------------------------
// One 16-row M-tile of the block-gate GEMM  S' = S * B  (complex) via WMMA.
// elem(row, s) = (row>>rsh)*hiStride + (row&rmask)*loStride + s*sStride
//   blk3: rsh=8 hi=0   rmask=255 lo=16 sS=1
//   blk2: rsh=4 hi=256 rmask=15  lo=1  sS=16
//   blk1/blk0-tile: rsh=8 hi=0 rmask=255 lo=1 sS=256
// ---------------------------------------------------------------------------
static __device__ __forceinline__ void apply_mtile(
    float* lre, float* lim, const GateFrag& f,
    int hiStride, int rsh, int rmask, int loStride, int sStride,
    int mt, int lane)
{
  const int khalf = (lane >> 4) << 1;
  const int row   = mt * 16 + (lane & 15);
  const int abase = (row >> rsh) * hiStride + (row & rmask) * loStride;
  v8f cr = {}; v8f ci = {};
#pragma unroll
  for (int ks = 0; ks < 4; ++ks) {
    const int s0 = ks * 4 + khalf;
    v2f ar, ai;
    ar.x = lre[abase + s0 * sStride]; ar.y = lre[abase + (s0 + 1) * sStride];
    ai.x = lim[abase + s0 * sStride]; ai.y = lim[abase + (s0 + 1) * sStride];
    cr = wmma_f32(ar, f.br[ks], cr);   // + Sr*Br
    cr = wmma_f32(ai, f.bn[ks], cr);   // - Si*Bi
    ci = wmma_f32(ar, f.bi[ks], ci);   // + Sr*Bi
    ci = wmma_f32(ai, f.br[ks], ci);   // + Si*Br
  }
  const int cb = (lane & 15) * sStride;      // output column
#pragma unroll
  for (int v = 0; v < 8; ++v) {
    const int rd   = mt * 16 + v + 8 * (lane >> 4);
    const int addr = (rd >> rsh) * hiStride + (rd & rmask) * loStride + cb;
    lre[addr] = cr[v];
    lim[addr] = ci[v];
  }
}

static __device__ __forceinline__ void dqc_apply_gate(
    float* lre, float* lim, const GateFrag& f,
    int hiStride, int rsh, int rmask, int loStride, int sStride,
    int wave, int lane)
{
  // Exactly 2 M-tiles per wave (16 tiles / 8 waves): fully uniform, EXEC stays
  // all-ones around every WMMA (ISA requirement), no divergent loop structure.
  apply_mtile(lre, lim, f, hiStride, rsh, rmask, loStride, sStride, wave,     lane);
  apply_mtile(lre, lim, f, hiStride, rsh, rmask, loStride, sStride, wave + 8, lane);
}

// ---------------------------------------------------------------------------
// Kernel 2: simulate one batch element per workgroup (256 thr = 8 waves).
// State (re/im planes, 64K floats each) lives in a per-slot global slice;
// gate applications stream 32KB tiles through LDS.
// ---------------------------------------------------------------------------
__global__ void __launch_bounds__(256)
dqc_simulate(const float* __restrict__ x, const float* __restrict__ gates,
             float* __restrict__ out, float* __restrict__ state_base, int bstart) {
  const int b    = bstart + blockIdx.x;
  float* gre = state_base + (size_t)blockIdx.x * 131072;  // 65536 re + 65536 im
  float* gim = gre + 65536;

  __shared__ __align__(16) float lre[4096];
  __shared__ __align__(16) float lim[4096];
  __shared__ float ph[256], pl[256];

  const int tid  = threadIdx.x;
  const int wave = tid >> 5;
  const int lane = tid & 31;

  // ---- Encoding layer: product state |psi> = kron_q [cos((q+1)phi), sin((q+1)phi)]
  const float phi = acosf(atanf(x[b]));
  {
    float p = 1.0f;
#pragma unroll
    for (int j = 0; j < 8; ++j) {               // bit j of hi-byte <-> half-angle (8-j)*phi
      const float ang = (float)(8 - j) * phi;
      p *= ((tid >> j) & 1) ? sinf(ang) : cosf(ang);
    }
    ph[tid] = p;
    p = 1.0f;
#pragma unroll
    for (int j = 0; j < 8; ++j) {               // bit j of lo-byte <-> half-angle (16-j)*phi
      const float ang = (float)(16 - j) * phi;
      p *= ((tid >> j) & 1) ? sinf(ang) : cosf(ang);
    }
    pl[tid] = p;
  }
  __syncthreads();
  const float plv = pl[tid];
#pragma unroll 4
  for (int i = 0; i < 256; ++i) {
    gre[i * 256 + tid] = ph[i] * plv;
    gim[i * 256 + tid] = 0.0f;
  }
  __threadfence_block();
  __syncthreads();

  // ---- 8 layers
#pragma unroll 1
  for (int l = 0; l < 8; ++l) {
    // Gate B-fragments are tile-invariant: load once per layer, keep in VGPRs.
    const GateFrag g1 = load_gate(gates + (l * 4 + 1) * 768, lane);
    const GateFrag g2 = load_gate(gates + (l * 4 + 2) * 768, lane);
    const GateFrag g3 = load_gate(gates + (l * 4 + 3) * 768, lane);

    // Sub-pass 1: tiles = contiguous 4096 amps (bits 11:0); apply blk1,blk2,blk3.
#pragma unroll 1
    for (int tt = 0; tt < 16; ++tt) {
      // Fold previous layer's pending CNOT ctrl-bit12 -> flip bit11 into the load.
      const int lmask = (l > 0 && (tt & 1)) ? 0x800 : 0;
#pragma unroll
      for (int v = 0; v < 4; ++v) {
        const int e = 4 * tid + 1024 * v;
        const int se = e ^ lmask;
        *reinterpret_cast<float4*>(&lre[e]) =
            *reinterpret_cast<const float4*>(&gre[tt * 4096 + se]);
        *reinterpret_cast<float4*>(&lim[e]) =
            *reinterpret_cast<const float4*>(&gim[tt * 4096 + se]);
      }
      if (tt < 15) __builtin_prefetch(&gre[(tt + 1) * 4096 + 4 * tid], 0, 1);
      __syncthreads();
      dqc_apply_gate(lre, lim, g1, 0, 8, 255, 1, 256, wave, lane);
      __syncthreads();
      dqc_apply_gate(lre, lim, g2, 256, 4, 15, 1, 16, wave, lane);
      __syncthreads();
      dqc_apply_gate(lre, lim, g3, 0, 8, 255, 16, 1, wave, lane);
      __syncthreads();
      // Store with cross CNOTs (ctrl bit8 -> flip bit7), (ctrl bit4 -> flip bit3).
#pragma unroll
      for (int v = 0; v < 4; ++v) {
        const int e  = 4 * tid + 1024 * v;
        const int se = e ^ ((e & 0x100) ? 0x80 : 0) ^ ((e & 0x10) ? 0x8 : 0);
        *reinterpret_cast<float4*>(&gre[tt * 4096 + e]) =
            *reinterpret_cast<const float4*>(&lre[se]);
        *reinterpret_cast<float4*>(&gim[tt * 4096 + e]) =
            *reinterpret_cast<const float4*>(&lim[se]);
      }
      __threadfence_block();
      __syncthreads();
    }

    // Sub-pass 2: tiles = bits[15:12] x bits[7:0]; apply blk0 (bits 15:12).
    const GateFrag g0 = load_gate(gates + (l * 4 + 0) * 768, lane);
#pragma unroll 1
    for (int t2 = 0; t2 < 16; ++t2) {
#pragma unroll
      for (int v = 0; v < 4; ++v) {
        const int e = 4 * tid + 1024 * v;                 // lds layout [s=e>>8][r=e&255]
        const int g = (e >> 8) * 4096 + t2 * 256 + (e & 255);
        *reinterpret_cast<float4*>(&lre[e]) = *reinterpret_cast<const float4*>(&gre[g]);
        *reinterpret_cast<float4*>(&lim[e]) = *reinterpret_cast<const float4*>(&gim[g]);
      }
      __syncthreads();
      dqc_apply_gate(lre, lim, g0, 0, 8, 255, 1, 256, wave, lane);
      __syncthreads();
#pragma unroll
      for (int v = 0; v < 4; ++v) {
        const int e = 4 * tid + 1024 * v;
        const int g = (e >> 8) * 4096 + t2 * 256 + (e & 255);
        *reinterpret_cast<float4*>(&gre[g]) = *reinterpret_cast<const float4*>(&lre[e]);
        *reinterpret_cast<float4*>(&gim[g]) = *reinterpret_cast<const float4*>(&lim[e]);
      }
      __threadfence_block();
      __syncthreads();
    }
  }

  // ---- Readout: sum_n |psi|^2 * (16 - 2*popcount(sigma(n))), sigma = pending
  // last-layer CNOT (ctrl bit12 -> flip bit11).
  float acc = 0.0f;
#pragma unroll 4
  for (int i = 0; i < 256; ++i) {
    const int n = i * 256 + tid;
    const float re = gre[n], im = gim[n];
    const int nn = n ^ ((n & 0x1000) ? 0x800 : 0);
    acc += (re * re + im * im) * (16.0f - 2.0f * (float)__popc(nn));
  }
  lre[tid] = acc;
  __syncthreads();
#pragma unroll 1
  for (int off = 128; off > 0; off >>= 1) {
    if (tid < off) lre[tid] += lre[tid + off];
    __syncthreads();
  }
  if (tid == 0) out[b] = lre[0];
}

// ---------------------------------------------------------------------------
// Host launch: build gates once, then simulate the batch in ws-sized chunks.
// ---------------------------------------------------------------------------
extern "C" void kernel_launch(void* const* d_in, const int* in_sizes, int n_in,
                              void* d_out, int out_size, void* d_ws, size_t ws_size,
                              hipStream_t stream) {
  const float* x      = (const float*)d_in[0];   // (512, 1) f32
  const float* params = (const float*)d_in[1];   // (8, 3, 16) f32
  float* out = (float*)d_out;                    // (512, 1) f32
  float* ws  = (float*)d_ws;

  float* gates      = ws;           // 8*4*768 = 24576 floats
  float* state_base = ws + 32768;   // 128 KB offset; 131072 floats per slot

  dqc_build_gates<<<dim3(8, 4), 256, 0, stream>>>(params, gates);

  const long slotFloats = 131072;   // 512 KB per batch element (re+im)
  long avail = ((long)(ws_size / 4) - 32768) / slotFloats;
  if (avail < 1) avail = 1;
  int chunk = (avail > 512) ? 512 : (int)avail;

  for (int s = 0; s < 512; s += chunk) {
    int n = 512 - s;
    if (n > chunk) n = chunk;
    dqc_simulate<<<n, 256, 0, stream>>>(x, gates, out, state_base, s);
  }
}